// AttentionSocialPooling_43696997270038
// MI455X (gfx1250) — compile-verified
//
#include <hip/hip_runtime.h>
#include <math.h>

typedef float v2f __attribute__((ext_vector_type(2)));
typedef float v8f __attribute__((ext_vector_type(8)));

#define NAGT 128          // N agents
#define AHID 16           // hidden units
#define RADIUS2 2500.0f   // 50^2

// Single-instruction ReLU: v_med3_f32(x, 0, +inf) clamps to [0, inf).
__device__ __forceinline__ float relu1(float x) {
    return __builtin_amdgcn_fmed3f(x, 0.0f, __builtin_inff());
}

// Merge two registers whose per-row partial sums are spread across lanes:
// lanes with (l15 & s)==0 end up holding a's row set, others b's, each
// reduced pairwise over lane distance s. One ds_bpermute per merge.
__device__ __forceinline__ float merge_rows(float a, float b, int bit, int s) {
    float t = bit ? a : b;          // send the value the partner needs
    float u = __shfl_xor(t, s, 32);
    return (bit ? b : a) + u;
}

// One block = 256 threads = 8 waves, handles 16 i's of one (b,t).
// grid.x = B*T, grid.y = 8 (i-tile). Each wave: 2 i's x 8 chunks of 16 j's,
// one v_wmma_f32_16x16x4_f32 per chunk (K=4 == the 4 MLP input features).
__global__ __launch_bounds__(256) void asp_wmma_kernel(
    const float* __restrict__ positions,  // (B,T,N,2)
    const float* __restrict__ W1,         // (4,16) row-major
    const float* __restrict__ b1,         // (16)
    const float* __restrict__ W2,         // (16,1)
    const float* __restrict__ b2,         // (1)
    float* __restrict__ out)              // (B,T,N,2)
{
    __shared__ float sp[NAGT * 2];

    const int bt   = blockIdx.x;
    const int tid  = threadIdx.x;
    const int lane = tid & 31;
    const int wave = tid >> 5;
    const int half = lane >> 4;   // 0: K={0,1} slice, 1: K={2,3} slice
    const int l15  = lane & 15;

    // Stage this (b,t)'s positions (1 KB) into LDS, shared by all 8 waves.
    const float* pbase = positions + (size_t)bt * NAGT * 2;
    sp[tid] = pbase[tid];
    __syncthreads();

    // B matrix = W1 (4x16), K split across lane halves per ISA B layout.
    const int k0 = half * 2;
    v2f bm;
    bm.x = W1[(k0 + 0) * AHID + l15];
    bm.y = W1[(k0 + 1) * AHID + l15];
    const float b1n = b1[l15];
    const float w2n = W2[l15];
    const float b2s = b2[0];

    // Loop-invariant C accumulator: C[m][n] = b1[n] (lane splat), folded
    // into the WMMA so the bias add is free. Hoisted so it stays pinned in
    // 8 VGPRs instead of being re-mov'd every chunk.
    const v8f cacc = {b1n, b1n, b1n, b1n, b1n, b1n, b1n, b1n};

    // Preload the 8 chunk j-positions for this lane's slot (i-independent).
    float pjx[8], pjy[8];
    #pragma unroll
    for (int c = 0; c < 8; ++c) {
        pjx[c] = sp[(c * 16 + l15) * 2 + 0];
        pjy[c] = sp[(c * 16 + l15) * 2 + 1];
    }

    const int   mrow   = (l15 & 7) + half * 8;    // row this lane finalizes
    const float active = (l15 < 8) ? 1.0f : 0.0f; // mask duplicate lanes

    #pragma unroll
    for (int ii = 0; ii < 2; ++ii) {
        const int i = blockIdx.y * 16 + wave * 2 + ii;
        const float pix = sp[i * 2 + 0];
        const float piy = sp[i * 2 + 1];

        float accx = 0.0f, accy = 0.0f, acccnt = 0.0f;

        #pragma unroll
        for (int c = 0; c < 8; ++c) {
            // Both lane halves compute diff for j = c*16 + l15.
            const float dx = pjx[c] - pix;
            const float dy = pjy[c] - piy;
            const float d2 = dx * dx + dy * dy;
            const float mk = (d2 > 0.0f && d2 < RADIUS2) ? 1.0f : 0.0f;

            // A matrix (16x4 f32): lanes 0-15 hold K0,K1 = pos_i;
            // lanes 16-31 hold K2,K3 = diff of row m = l15.
            v2f am;
            am.x = half ? dx : pix;
            am.y = half ? dy : piy;

            v8f d = __builtin_amdgcn_wmma_f32_16x16x4_f32(
                false, am, false, bm, (short)0, cacc, false, false);

            // Second layer pointwise: p_r = relu(h[row r][n]) * W2[n].
            float p0 = relu1(d[0]) * w2n, p1 = relu1(d[1]) * w2n;
            float p2 = relu1(d[2]) * w2n, p3 = relu1(d[3]) * w2n;
            float p4 = relu1(d[4]) * w2n, p5 = relu1(d[5]) * w2n;
            float p6 = relu1(d[6]) * w2n, p7 = relu1(d[7]) * w2n;

            // Batched reduction: 8 regs x 16 lanes -> 1 reg, row (l15&7)+8h.
            const int bit0 = l15 & 1, bit1 = l15 & 2, bit2 = l15 & 4;
            float q0 = merge_rows(p0, p1, bit0, 1);
            float q1 = merge_rows(p2, p3, bit0, 1);
            float q2 = merge_rows(p4, p5, bit0, 1);
            float q3 = merge_rows(p6, p7, bit0, 1);
            float r0 = merge_rows(q0, q1, bit1, 2);
            float r1 = merge_rows(q2, q3, bit1, 2);
            float s0 = merge_rows(r0, r1, bit2, 4);
            float full = s0 + __shfl_xor(s0, 8, 32);

            // One sigmoid per lane covers all 16 rows (x2 duplication).
            const float att = 1.0f / (1.0f + __expf(-(full + b2s)));

            // Gather diff/mask of the row this lane finalizes (lane mrow).
            const float dmx = __shfl(dx, mrow, 32);
            const float dmy = __shfl(dy, mrow, 32);
            const float mkm = __shfl(mk, mrow, 32);

            const float w = att * mkm * active;
            accx   += dmx * w;
            accy   += dmy * w;
            acccnt += mkm * active;
        }

        // Distinct rows live in lanes 0-7 (rows 0-7) and 16-23 (rows 8-15);
        // lanes 8-15/24-31 hold zeros, so the xor-8 stage is unnecessary:
        // reduce with strides 16, then 1,2,4.
        accx   += __shfl_xor(accx,   16, 32);
        accy   += __shfl_xor(accy,   16, 32);
        acccnt += __shfl_xor(acccnt, 16, 32);
        #pragma unroll
        for (int s = 1; s <= 4; s <<= 1) {
            accx   += __shfl_xor(accx,   s, 32);
            accy   += __shfl_xor(accy,   s, 32);
            acccnt += __shfl_xor(acccnt, s, 32);
        }

        if (lane == 0) {
            const float inv = 1.0f / fmaxf(acccnt, 1e-6f);
            float* o = out + ((size_t)bt * NAGT + i) * 2;
            o[0] = accx * inv;
            o[1] = accy * inv;
        }
    }
}

extern "C" void kernel_launch(void* const* d_in, const int* in_sizes, int n_in,
                              void* d_out, int out_size, void* d_ws, size_t ws_size,
                              hipStream_t stream) {
    (void)n_in; (void)out_size; (void)d_ws; (void)ws_size;
    const float* positions = (const float*)d_in[0];
    const float* W1        = (const float*)d_in[1];
    const float* b1        = (const float*)d_in[2];
    const float* W2        = (const float*)d_in[3];
    const float* b2        = (const float*)d_in[4];
    float* out             = (float*)d_out;

    const int BT = in_sizes[0] / (NAGT * 2);   // B*T = 512
    dim3 grid(BT, 8, 1);
    dim3 block(256, 1, 1);
    asp_wmma_kernel<<<grid, block, 0, stream>>>(positions, W1, b1, W2, b2, out);
}